// GLiNER_20624432956210
// MI455X (gfx1250) — compile-verified
//
#include <hip/hip_runtime.h>

// ---------------- problem constants (from setup_inputs) ----------------
constexpr int B_   = 8;
constexpr int L_   = 1024;
constexpr int H_   = 768;
constexpr int E_   = 16;
constexpr int W_   = 504;
constexpr int DFF  = 512;
constexpr int MAXSW = 12;
constexpr int NSP  = 5982;            // total spans
constexpr int NE   = NSP * E_;        // 95712 candidates per batch
constexpr unsigned NP = 131072;       // next pow2 >= NE (sort size)

typedef float v2f __attribute__((ext_vector_type(2)));
typedef float v8f __attribute__((ext_vector_type(8)));
typedef int   v4i __attribute__((ext_vector_type(4)));

#define GLOBAL_AS __attribute__((address_space(1)))
#define SHARED_AS __attribute__((address_space(3)))

#if __has_builtin(__builtin_amdgcn_global_load_async_to_lds_b128)
#define HAVE_ASYNC_LDS 1
#else
#define HAVE_ASYNC_LDS 0
#endif

__device__ __forceinline__ void async_wait_all() {
#if __has_builtin(__builtin_amdgcn_s_wait_asynccnt)
    __builtin_amdgcn_s_wait_asynccnt(0);
#else
    asm volatile("s_wait_asynccnt 0" ::: "memory");
#endif
}

// ---------------- span table ----------------
__global__ void init_spans_k(int* ss, int* ee) {
    if (threadIdx.x == 0 && blockIdx.x == 0) {
        int n = 0;
        for (int s = 0; s < W_; ++s) {
            int emax = (s + MAXSW - 1 < W_ - 1) ? s + MAXSW - 1 : W_ - 1;
            for (int e = s; e <= emax; ++e) { ss[n] = s; ee[n] = e; ++n; }
        }
    }
}

// ---------------- word pooling ----------------
__global__ void word_init_k(int* wfirst, int* wcnt) {
    int t = blockIdx.x * 256 + threadIdx.x;
    if (t < B_ * W_) { wfirst[t] = 0x7fffffff; wcnt[t] = 0; }
}

__global__ void word_scan_k(const int* __restrict__ tmask, const int* __restrict__ widx,
                            int* wfirst, int* wcnt) {
    int t = blockIdx.x * 256 + threadIdx.x;
    if (t >= B_ * L_) return;
    int b = t / L_, l = t % L_;
    int w = widx[t];
    if (tmask[t] == 1 && w >= 0) {
        atomicMin(&wfirst[b * W_ + w], l);
        atomicAdd(&wcnt[b * W_ + w], 1);
    }
}

__global__ void word_pool_k(const float* __restrict__ hs, const int* __restrict__ wfirst,
                            const int* __restrict__ wcnt, float* __restrict__ we) {
    size_t t = (size_t)blockIdx.x * 256 + threadIdx.x;
    if (t >= (size_t)B_ * W_ * H_) return;
    int h = (int)(t % H_);
    size_t bw = t / H_;
    int b = (int)(bw / W_);
    int cnt = wcnt[bw], f = wfirst[bw];
    float s = 0.f;
    for (int i = 0; i < cnt; ++i) s += hs[((size_t)b * L_ + f + i) * H_ + h];
    we[t] = s / (float)(cnt > 1 ? cnt : 1);
}

// ---------------- entity side (tiny: plain VALU) ----------------
__global__ void ent_pos_k(const int* __restrict__ emask, int* pos) {
    int b = blockIdx.x;
    if (threadIdx.x == 0) {
        int c = 0;
        for (int l = 0; l < L_ && c < E_; ++l)
            if (emask[(size_t)b * L_ + l] == 1) pos[b * E_ + c++] = l;
    }
}

__global__ void ent_mlp1_k(const float* __restrict__ hs, const int* __restrict__ pos,
                           const float* __restrict__ w1, const float* __restrict__ b1,
                           float* __restrict__ a1) {
    int b = blockIdx.x;
    for (int o = threadIdx.x; o < E_ * DFF; o += blockDim.x) {
        int ei = o / DFF, d = o % DFF;
        const float* tok = hs + ((size_t)b * L_ + pos[b * E_ + ei]) * H_;
        float s = b1[d];
        for (int h = 0; h < H_; ++h) s += tok[h] * w1[(size_t)h * DFF + d];
        a1[((size_t)b * E_ + ei) * DFF + d] = s > 0.f ? s : 0.01f * s;
    }
}

__global__ void ent_mlp2_k(const float* __restrict__ a1, const float* __restrict__ w2,
                           const float* __restrict__ b2, float* __restrict__ ev) {
    int b = blockIdx.x;
    for (int o = threadIdx.x; o < E_ * H_; o += blockDim.x) {
        int ei = o / H_, h = o % H_;
        const float* ar = a1 + ((size_t)b * E_ + ei) * DFF;
        float s = b2[h];
        for (int d = 0; d < DFF; ++d) s += ar[d] * w2[(size_t)d * H_ + h];
        ev[((size_t)b * E_ + ei) * H_ + h] = s;
    }
}

// M2t[b,e,d] = sum_h spanW2[d,h]*ent_vec[b,e,h]  (TRANSPOSED: e-major, so WMMA B-fragment
// pairs {M2[k0],M2[k0+1]} are contiguous 8B) ; c[b,e] = sum_h spanB2[h]*ent_vec[b,e,h]
__global__ void m2c_k(const float* __restrict__ sw2, const float* __restrict__ sb2,
                      const float* __restrict__ ev, float* __restrict__ M2t,
                      float* __restrict__ cvec) {
    int b = blockIdx.x;
    const float* evb = ev + (size_t)b * E_ * H_;
    for (int o = threadIdx.x; o < DFF * E_; o += blockDim.x) {
        int d = o / E_, e = o % E_;
        const float* wr  = sw2 + (size_t)d * H_;
        const float* evr = evb + (size_t)e * H_;
        float s = 0.f;
        for (int h = 0; h < H_; ++h) s += wr[h] * evr[h];
        M2t[(size_t)b * E_ * DFF + (size_t)e * DFF + d] = s;
    }
    if (threadIdx.x < E_) {
        const float* evr = evb + (size_t)threadIdx.x * H_;
        float s = 0.f;
        for (int h = 0; h < H_; ++h) s += sb2[h] * evr[h];
        cvec[b * E_ + threadIdx.x] = s;
    }
}

// ---------------- WMMA GEMM:  [P|Q] = word_emb(504x768) @ W1(768x1024) per batch ----------
// Block: 128 threads = 4 waves; 64x64 tile; LDS-staged K chunks of 32 (async direct-to-LDS
// copies when available); f32 WMMA 16x16x4. LDS rows padded (36 / 68 floats) to keep
// 16B-aligned b128 staging while spreading banks.
constexpr int APAD = 36;   // A tile row stride (floats), 64 rows x 32 cols
constexpr int BPAD = 68;   // B tile row stride (floats), 32 rows x 64 cols

__global__ void gemm_pq_k(const float* __restrict__ we, const float* __restrict__ w1,
                          float* __restrict__ P, float* __restrict__ Q) {
    __shared__ float As[64 * APAD];
    __shared__ float Bs[32 * BPAD];
    int mt = blockIdx.x;              // 8 M-tiles over 504
    int nt = blockIdx.y;              // 16 N-tiles over 1024 (first 8 -> P, last 8 -> Q)
    int b  = blockIdx.z;
    const float* A = we + (size_t)b * W_ * H_;
    bool isQ = nt >= 8;
    const float* Bm = w1 + (isQ ? (size_t)H_ * DFF : 0);     // [768,512] half
    int ncol0 = (nt - (isQ ? 8 : 0)) * 64;
    float* Cout = (isQ ? Q : P) + (size_t)b * W_ * DFF;

    int tid = threadIdx.x;
    int wv = tid >> 5, lane = tid & 31;
    int mrow = lane & 15, kh = lane >> 4, ncol = lane & 15;

    v8f acc[4] = {v8f{}, v8f{}, v8f{}, v8f{}};
    for (int kc = 0; kc < H_; kc += 32) {
        __syncthreads();              // previous chunk's LDS reads done
        // stage A: 64x32 floats = 512 x 16B
        #pragma unroll
        for (int i = 0; i < 4; ++i) {
            int idx = tid + i * 128;
            int r = idx >> 3, c4 = (idx & 7) << 2;
            int gr = mt * 64 + r;
            float* dst = &As[r * APAD + c4];
            if (gr < W_) {
#if HAVE_ASYNC_LDS
                __builtin_amdgcn_global_load_async_to_lds_b128(
                    (GLOBAL_AS v4i*)(A + (size_t)gr * H_ + kc + c4),
                    (SHARED_AS v4i*)dst, 0, 0);
#else
                *(float4*)dst = *(const float4*)(A + (size_t)gr * H_ + kc + c4);
#endif
            } else {
                *(float4*)dst = make_float4(0.f, 0.f, 0.f, 0.f);
            }
        }
        // stage B: 32x64 floats = 512 x 16B
        #pragma unroll
        for (int i = 0; i < 4; ++i) {
            int idx = tid + i * 128;
            int r = idx >> 4, c4 = (idx & 15) << 2;
            float* dst = &Bs[r * BPAD + c4];
#if HAVE_ASYNC_LDS
            __builtin_amdgcn_global_load_async_to_lds_b128(
                (GLOBAL_AS v4i*)(Bm + (size_t)(kc + r) * DFF + ncol0 + c4),
                (SHARED_AS v4i*)dst, 0, 0);
#else
            *(float4*)dst = *(const float4*)(Bm + (size_t)(kc + r) * DFF + ncol0 + c4);
#endif
        }
#if HAVE_ASYNC_LDS
        async_wait_all();             // my copies landed in LDS
#endif
        __syncthreads();              // everyone's copies visible
        if (kc + 32 < H_) {           // prefetch next chunk -> global_prefetch_b8
            int gr = mt * 64 + (tid >> 1);
            if (gr < W_) __builtin_prefetch(A + (size_t)gr * H_ + kc + 32, 0, 1);
            __builtin_prefetch(Bm + (size_t)(kc + 32 + (tid & 31)) * DFF + ncol0, 0, 1);
        }
        #pragma unroll
        for (int kk = 0; kk < 8; ++kk) {
            int k0 = kk * 4 + kh * 2;
            v2f a = *(const v2f*)(&As[(wv * 16 + mrow) * APAD + k0]);
            #pragma unroll
            for (int j = 0; j < 4; ++j) {
                v2f bb;
                bb.x = Bs[k0 * BPAD + j * 16 + ncol];
                bb.y = Bs[(k0 + 1) * BPAD + j * 16 + ncol];
                acc[j] = __builtin_amdgcn_wmma_f32_16x16x4_f32(
                    false, a, false, bb, (short)0, acc[j], false, false);
            }
        }
    }
    #pragma unroll
    for (int j = 0; j < 4; ++j)
        #pragma unroll
        for (int v = 0; v < 8; ++v) {
            int gr = mt * 64 + wv * 16 + v + 8 * kh;
            if (gr < W_) Cout[(size_t)gr * DFF + ncol0 + j * 16 + ncol] = acc[j][v];
        }
}

// ---------------- fused span scoring: logits = leaky(P[s]+Q[e]+b1) @ M2 + c ------------
// Block 128 = 4 waves; each wave scores 16 spans x 16 entities via 128 f32 WMMAs (K=512).
// M2t (e-major) staged in LDS with padded stride so each B fragment is one ds_load_b64.
constexpr int MPAD = 516;  // 512 + 4: 16B-aligned rows, distinct bank offset per row

__global__ void span_logits_k(const float* __restrict__ P, const float* __restrict__ Q,
                              const float* __restrict__ b1, const float* __restrict__ M2t,
                              const float* __restrict__ cvec,
                              const int* __restrict__ ss, const int* __restrict__ ee,
                              float* __restrict__ logits) {
    __shared__ float Ms[E_ * MPAD];
    int b = blockIdx.y;
    int tid = threadIdx.x;
    const float* M2b = M2t + (size_t)b * E_ * DFF;   // [16][512]
    #pragma unroll
    for (int i = 0; i < 16; ++i) {                   // 2048 x 16B over 128 threads
        int idx = tid + i * 128;
        int e = idx >> 7, c4 = (idx & 127) << 2;
        *(float4*)(&Ms[e * MPAD + c4]) = *(const float4*)(M2b + (size_t)e * DFF + c4);
    }
    __syncthreads();

    int wv = tid >> 5, lane = tid & 31;
    int sb = blockIdx.x * 64 + wv * 16;
    if (sb >= NSP) return;                           // uniform per wave, after barrier
    int mrow = lane & 15, kh = lane >> 4, ne = lane & 15;
    int n = sb + mrow;
    int ncl = n < NSP ? n : NSP - 1;
    int s = ss[ncl], e = ee[ncl];
    const float* Pp = P + ((size_t)b * W_ + s) * DFF;
    const float* Qp = Q + ((size_t)b * W_ + e) * DFF;

    v8f acc = v8f{};
    #pragma unroll 4
    for (int kk = 0; kk < 128; ++kk) {
        int k0 = kk * 4 + kh * 2;
        v2f p  = *(const v2f*)(Pp + k0);
        v2f q  = *(const v2f*)(Qp + k0);
        v2f bv = *(const v2f*)(b1 + k0);
        float x0 = p.x + q.x + bv.x;
        float x1 = p.y + q.y + bv.y;
        v2f a;
        a.x = x0 > 0.f ? x0 : 0.01f * x0;            // LeakyReLU(0.01)
        a.y = x1 > 0.f ? x1 : 0.01f * x1;
        v2f bb = *(const v2f*)(&Ms[ne * MPAD + k0]); // single ds_load_b64
        acc = __builtin_amdgcn_wmma_f32_16x16x4_f32(
            false, a, false, bb, (short)0, acc, false, false);
    }
    float cadd = cvec[b * E_ + ne];
    #pragma unroll
    for (int v = 0; v < 8; ++v) {
        int nn = sb + v + 8 * kh;
        if (nn < NSP)
            logits[((size_t)b * NSP + nn) * E_ + ne] = acc[v] + cadd;
    }
}

// ---------------- sort keys: ascending u64 sort == stable descending-prob argsort ------
__global__ void make_keys_k(const float* __restrict__ logits,
                            unsigned long long* __restrict__ keys,
                            float* __restrict__ sel) {
    size_t t = (size_t)blockIdx.x * 256 + threadIdx.x;
    if (t >= (size_t)B_ * NP) return;
    int b = (int)(t >> 17);
    unsigned i = (unsigned)(t & (NP - 1));
    unsigned long long key;
    if (i < (unsigned)NE) {
        float f = logits[(size_t)b * NE + i];
        unsigned u = __float_as_uint(f);
        unsigned m = (u >> 31) ? ~u : (u | 0x80000000u);  // ascending float order
        unsigned k32 = ~m;                                 // descending
        key = ((unsigned long long)k32 << 32) | (unsigned long long)i;
        sel[(size_t)b * NE + i] = 0.f;                     // init selected output
    } else {
        key = ~0ull;                                       // pad sorts last
    }
    keys[(size_t)b * NP + i] = key;
}

__global__ void bitonic_step_k(unsigned long long* __restrict__ keys,
                               unsigned j, unsigned k) {
    size_t t = (size_t)blockIdx.x * 256 + threadIdx.x;
    if (t >= (size_t)B_ * NP) return;
    unsigned b = (unsigned)(t >> 17);
    unsigned i = (unsigned)(t & (NP - 1));
    unsigned ixj = i ^ j;
    if (ixj > i) {
        unsigned long long* seg = keys + (size_t)b * NP;
        unsigned long long a = seg[i], c = seg[ixj];
        bool up = (i & k) == 0;
        if ((a > c) == up) { seg[i] = c; seg[ixj] = a; }
    }
}

// ---------------- greedy NMS decode (serial tail; breaks at first logit<=0) ------------
__global__ void greedy_decode_k(const unsigned long long* __restrict__ keys,
                                const float* __restrict__ logits,
                                const int* __restrict__ ss, const int* __restrict__ ee,
                                float* __restrict__ sel) {
    int b = threadIdx.x;
    if (b >= B_) return;
    const unsigned long long* seg = keys + (size_t)b * NP;
    unsigned long long cov[8] = {0, 0, 0, 0, 0, 0, 0, 0};
    for (unsigned t = 0; t < NP; ++t) {
        unsigned long long key = seg[t];
        unsigned idx = (unsigned)(key & 0xffffffffull);
        if (idx >= (unsigned)NE) break;                    // padding region
        float lg = logits[(size_t)b * NE + idx];
        if (!(lg > 0.f)) break;                            // sigmoid<=0.5, sorted -> done
        int n = (int)(idx / E_);
        int s = ss[n], e = ee[n];
        int w0 = s >> 6, w1 = e >> 6;
        unsigned long long m0, m1 = 0ull;
        if (w0 == w1) {
            m0 = (((1ull << (e - s + 1)) - 1ull)) << (s & 63);
        } else {
            m0 = (~0ull) << (s & 63);
            m1 = (1ull << ((e & 63) + 1)) - 1ull;
        }
        bool ov = (cov[w0] & m0) != 0ull || (w1 != w0 && (cov[w1] & m1) != 0ull);
        if (!ov) {
            cov[w0] |= m0;
            if (w1 != w0) cov[w1] |= m1;
            sel[(size_t)b * NE + idx] = 1.f;
        }
    }
}

// ---------------- launcher ----------------
extern "C" void kernel_launch(void* const* d_in, const int* in_sizes, int n_in,
                              void* d_out, int out_size, void* d_ws, size_t ws_size,
                              hipStream_t stream) {
    const float* hs   = (const float*)d_in[0];
    const float* ew1  = (const float*)d_in[1];
    const float* eb1  = (const float*)d_in[2];
    const float* ew2  = (const float*)d_in[3];
    const float* eb2  = (const float*)d_in[4];
    const float* sw1  = (const float*)d_in[5];
    const float* sb1  = (const float*)d_in[6];
    const float* sw2  = (const float*)d_in[7];
    const float* sb2  = (const float*)d_in[8];
    const int* tmask  = (const int*)d_in[9];
    const int* emask  = (const int*)d_in[10];
    const int* widx   = (const int*)d_in[11];

    float* out_logits = (float*)d_out;                       // [B, NSP, E]
    float* out_sel    = out_logits + (size_t)B_ * NE;        // [B, NSP, E]

    // workspace carve-out
    char* wp = (char*)d_ws;
    auto carve = [&](size_t bytes) -> void* {
        void* p = (void*)wp;
        wp += (bytes + 255) & ~(size_t)255;
        return p;
    };
    int*   span_s = (int*)carve((size_t)NSP * 4);
    int*   span_e = (int*)carve((size_t)NSP * 4);
    int*   wfirst = (int*)carve((size_t)B_ * W_ * 4);
    int*   wcnt   = (int*)carve((size_t)B_ * W_ * 4);
    int*   epos   = (int*)carve((size_t)B_ * E_ * 4);
    float* wemb   = (float*)carve((size_t)B_ * W_ * H_ * 4);
    float* a1     = (float*)carve((size_t)B_ * E_ * DFF * 4);
    float* evec   = (float*)carve((size_t)B_ * E_ * H_ * 4);
    float* M2t    = (float*)carve((size_t)B_ * E_ * DFF * 4);
    float* cvec   = (float*)carve((size_t)B_ * E_ * 4);
    float* P      = (float*)carve((size_t)B_ * W_ * DFF * 4);
    float* Q      = (float*)carve((size_t)B_ * W_ * DFF * 4);
    unsigned long long* keys = (unsigned long long*)carve((size_t)B_ * NP * 8);

    // 1) span table + word pooling metadata
    init_spans_k<<<1, 32, 0, stream>>>(span_s, span_e);
    word_init_k<<<(B_ * W_ + 255) / 256, 256, 0, stream>>>(wfirst, wcnt);
    word_scan_k<<<(B_ * L_ + 255) / 256, 256, 0, stream>>>(tmask, widx, wfirst, wcnt);
    word_pool_k<<<(int)(((size_t)B_ * W_ * H_ + 255) / 256), 256, 0, stream>>>(
        hs, wfirst, wcnt, wemb);

    // 2) entity branch (tiny) -> M2t, c
    ent_pos_k<<<B_, 32, 0, stream>>>(emask, epos);
    ent_mlp1_k<<<B_, 256, 0, stream>>>(hs, epos, ew1, eb1, a1);
    ent_mlp2_k<<<B_, 256, 0, stream>>>(a1, ew2, eb2, evec);
    m2c_k<<<B_, 256, 0, stream>>>(sw2, sb2, evec, M2t, cvec);

    // 3) P/Q GEMM (WMMA f32, async LDS staging)
    gemm_pq_k<<<dim3(8, 16, B_), 128, 0, stream>>>(wemb, sw1, P, Q);

    // 4) fused span-MLP + logit projection (WMMA f32)
    span_logits_k<<<dim3((NSP + 63) / 64, B_), 128, 0, stream>>>(
        P, Q, sb1, M2t, cvec, span_s, span_e, out_logits);

    // 5) keys (+ zero 'selected'), bitonic sort, greedy decode
    make_keys_k<<<(int)(((size_t)B_ * NP + 255) / 256), 256, 0, stream>>>(
        out_logits, keys, out_sel);
    for (unsigned k = 2; k <= NP; k <<= 1)
        for (unsigned j = k >> 1; j > 0; j >>= 1)
            bitonic_step_k<<<(int)(((size_t)B_ * NP + 255) / 256), 256, 0, stream>>>(
                keys, j, k);
    greedy_decode_k<<<1, 32, 0, stream>>>(keys, out_logits, span_s, span_e, out_sel);
}